// FakeReviewGNN_67826123539051
// MI455X (gfx1250) — compile-verified
//
#include <hip/hip_runtime.h>
#include <math.h>

#define NNODES 100000

typedef __attribute__((ext_vector_type(2))) float v2f;
typedef __attribute__((ext_vector_type(8))) float v8f;

// ---------------- utility kernels ----------------

__global__ void fill_kernel(float* __restrict__ p, float v, int n) {
  int i = blockIdx.x * blockDim.x + threadIdx.x;
  if (i < n) p[i] = v;
}

// deg[dst[e]] += 1 ; deg pre-initialized to 1.0 (self-loop)
__global__ void deg_kernel(const int* __restrict__ dst, float* __restrict__ deg, int E) {
  int e = blockIdx.x * blockDim.x + threadIdx.x;
  if (e < E) unsafeAtomicAdd(&deg[dst[e]], 1.0f);
}

// dinv = rsqrt(deg)  (deg >= 1 always due to self-loop)
__global__ void rsqrt_kernel(float* __restrict__ d, int n) {
  int i = blockIdx.x * blockDim.x + threadIdx.x;
  if (i < n) d[i] = rsqrtf(d[i]);
}

// pad x [N x 7] -> xp [N x 8] with zero column 7 (makes K a multiple of 4,
// and every A-fragment load an aligned b64 load)
__global__ void pad_x_kernel(const float* __restrict__ x, float* __restrict__ xp, int n) {
  int i = blockIdx.x * blockDim.x + threadIdx.x;   // over n*8
  if (i >= n * 8) return;
  const int r = i >> 3, c = i & 7;
  xp[i] = (c < 7) ? x[r * 7 + c] : 0.0f;
}

// pad W1 [7 x 64] -> wp [8 x 64] with zero row 7
__global__ void pad_w_kernel(const float* __restrict__ w, float* __restrict__ wp) {
  int i = blockIdx.x * blockDim.x + threadIdx.x;   // over 8*64
  if (i >= 8 * 64) return;
  wp[i] = (i < 7 * 64) ? w[i] : 0.0f;
}

// ---------------- dense GEMM via V_WMMA_F32_16X16X4_F32 ----------------
// Y[M x NOUT] = X[M x K] * W[K x NOUT], row-major, K % 4 == 0, M % 16 == 0.
// One wave computes a 16 x NOUT strip (all NOUT/16 tiles), fully unrolled,
// branch-free (EXEC all ones throughout, as WMMA requires).
// A layout (16x4 f32): lane L holds row m=L%16; VGPR j holds K = (L<16 ? j : 2+j)
// B layout (4x16 f32): lane L holds col n=L%16; VGPR j holds K = (L<16 ? j : 2+j)
// C/D layout: lane L, VGPR i -> (row = i + (L<16?0:8), col = L%16)
template <int K, int NOUT>
__global__ __launch_bounds__(32)
void gemm_wmma_kernel(const float* __restrict__ X, const float* __restrict__ W,
                      float* __restrict__ Y) {
  static_assert(K % 4 == 0 && NOUT % 16 == 0, "shape");
  constexpr int NT = NOUT / 16;
  const int lane = threadIdx.x;     // 0..31
  const int m    = lane & 15;
  const int half = lane >> 4;       // 0: K={0,1}, 1: K={2,3} within a k-step
  const int row  = blockIdx.x * 16 + m;
  const float* xrow = X + (size_t)row * K;

  v8f acc[NT] = {};

#pragma unroll
  for (int s = 0; s < K / 4; ++s) {
    const int kb = s * 4 + half * 2;
    const v2f a = *(const v2f*)(xrow + kb);          // aligned b64 load
#pragma unroll
    for (int t = 0; t < NT; ++t) {
      v2f b;
      b.x = W[(size_t)kb * NOUT + t * 16 + m];
      b.y = W[(size_t)(kb + 1) * NOUT + t * 16 + m];
      acc[t] = __builtin_amdgcn_wmma_f32_16x16x4_f32(
          /*neg_a=*/false, a, /*neg_b=*/false, b,
          /*c_mod=*/(short)0, acc[t], /*reuse_a=*/false, /*reuse_b=*/false);
    }
  }

#pragma unroll
  for (int t = 0; t < NT; ++t) {
#pragma unroll
    for (int i = 0; i < 8; ++i) {
      const int r = blockIdx.x * 16 + i + half * 8;
      Y[(size_t)r * NOUT + t * 16 + m] = acc[t][i];  // lanes coalesce over cols
    }
  }
}

// ---------------- edge scatter: out[dst] += h[src] * dinv[src]*dinv[dst] ----
// One thread per (edge, 4-feature chunk); native f32 atomics (L2-resident data).
__global__ void scatter_kernel(const float* __restrict__ h, const int* __restrict__ src,
                               const int* __restrict__ dst, const float* __restrict__ dinv,
                               float* __restrict__ out, int E, int F) {
  const long long idx = (long long)blockIdx.x * blockDim.x + threadIdx.x;
  const int chunks = F >> 2;
  const long long total = (long long)E * chunks;
  if (idx >= total) return;
  const int e  = (int)(idx / chunks);
  const int ch = (int)(idx % chunks);
  const int s = src[e];
  const int d = dst[e];
  const float w = dinv[s] * dinv[d];
  const float4 v = *(const float4*)(h + (size_t)s * F + ch * 4);
  float* o = out + (size_t)d * F + ch * 4;
  unsafeAtomicAdd(o + 0, v.x * w);
  unsafeAtomicAdd(o + 1, v.y * w);
  unsafeAtomicAdd(o + 2, v.z * w);
  unsafeAtomicAdd(o + 3, v.w * w);
}

// a = relu(a + h*dinv^2 (self-loop) + bias)
__global__ void finalize_kernel(float* __restrict__ a, const float* __restrict__ h,
                                const float* __restrict__ dinv, const float* __restrict__ bias,
                                int n, int F) {
  int i = blockIdx.x * blockDim.x + threadIdx.x;
  if (i >= n * F) return;
  const int r = i / F;
  const int f = i % F;
  const float di = dinv[r];
  const float v = a[i] + h[i] * di * di + bias[f];
  a[i] = v > 0.0f ? v : 0.0f;
}

// z = a[i,:32] @ Wc[32x2] + bc ; out = log_softmax(z)
__global__ void classifier_kernel(const float* __restrict__ a, const float* __restrict__ Wc,
                                  const float* __restrict__ bc, float* __restrict__ out, int n) {
  int i = blockIdx.x * blockDim.x + threadIdx.x;
  if (i >= n) return;
  const float* row = a + (size_t)i * 32;
  float z0 = bc[0], z1 = bc[1];
#pragma unroll
  for (int k = 0; k < 32; ++k) {
    const float v = row[k];
    z0 = fmaf(v, Wc[k * 2 + 0], z0);
    z1 = fmaf(v, Wc[k * 2 + 1], z1);
  }
  const float m = fmaxf(z0, z1);
  const float lse = m + logf(__expf(z0 - m) + __expf(z1 - m));
  out[(size_t)i * 2 + 0] = z0 - lse;
  out[(size_t)i * 2 + 1] = z1 - lse;
}

// ---------------- launch ----------------

extern "C" void kernel_launch(void* const* d_in, const int* in_sizes, int n_in,
                              void* d_out, int out_size, void* d_ws, size_t ws_size,
                              hipStream_t stream) {
  const float* x  = (const float*)d_in[0];
  const int*   ei = (const int*)d_in[1];
  const float* W1 = (const float*)d_in[2];
  const float* b1 = (const float*)d_in[3];
  const float* W2 = (const float*)d_in[4];
  const float* b2 = (const float*)d_in[5];
  const float* Wc = (const float*)d_in[6];
  const float* bc = (const float*)d_in[7];
  float* out = (float*)d_out;

  const int N = NNODES;               // 100000 = 6250 * 16 (no row padding needed)
  const int E = in_sizes[1] / 2;
  const int* src = ei;
  const int* dst = ei + E;

  // workspace (floats): dinv[N] | xp[N*8] | w1p[512] | h[N*64] | a[N*64]  (~55 MB)
  float* dinv = (float*)d_ws;
  float* xp   = dinv + N;
  float* w1p  = xp + (size_t)N * 8;
  float* h    = w1p + 512;                  // h1, later reused as h2 (N*32)
  float* a    = h + (size_t)N * 64;         // a1, later reused as a2 (N*32)

  const int T = 256;

  // --- symmetric normalization: dinv = rsqrt(1 + in_degree) ---
  fill_kernel<<<(N + T - 1) / T, T, 0, stream>>>(dinv, 1.0f, N);
  deg_kernel<<<(E + T - 1) / T, T, 0, stream>>>(dst, dinv, E);
  rsqrt_kernel<<<(N + T - 1) / T, T, 0, stream>>>(dinv, N);

  // --- zero-pad K: x [N,7] -> xp [N,8], W1 [7,64] -> w1p [8,64] ---
  pad_x_kernel<<<(N * 8 + T - 1) / T, T, 0, stream>>>(x, xp, N);
  pad_w_kernel<<<2, T, 0, stream>>>(W1, w1p);

  // --- layer 1: h1 = xp @ w1p (WMMA fp32), aggregate, relu ---
  gemm_wmma_kernel<8, 64><<<N / 16, 32, 0, stream>>>(xp, w1p, h);
  fill_kernel<<<(N * 64 + T - 1) / T, T, 0, stream>>>(a, 0.0f, N * 64);
  {
    const long long tot = (long long)E * (64 / 4);
    scatter_kernel<<<(unsigned)((tot + T - 1) / T), T, 0, stream>>>(h, src, dst, dinv, a, E, 64);
  }
  finalize_kernel<<<(N * 64 + T - 1) / T, T, 0, stream>>>(a, h, dinv, b1, N, 64);

  // --- layer 2: h2 = a1 @ W2 (WMMA fp32), aggregate, relu ---
  gemm_wmma_kernel<64, 32><<<N / 16, 32, 0, stream>>>(a, W2, h);
  fill_kernel<<<(N * 32 + T - 1) / T, T, 0, stream>>>(a, 0.0f, N * 32);
  {
    const long long tot = (long long)E * (32 / 4);
    scatter_kernel<<<(unsigned)((tot + T - 1) / T), T, 0, stream>>>(h, src, dst, dinv, a, E, 32);
  }
  finalize_kernel<<<(N * 32 + T - 1) / T, T, 0, stream>>>(a, h, dinv, b2, N, 32);

  // --- classifier + log_softmax ---
  classifier_kernel<<<(N + T - 1) / T, T, 0, stream>>>(a, Wc, bc, out, N);
}